// ResidualVirtualNode_60138132078773
// MI455X (gfx1250) — compile-verified
//
#include <hip/hip_runtime.h>
#include <hip/hip_bf16.h>
#include <stdint.h>

#define DIMH   512
#define DIMH2  1024
#define NGRAPH 512
#define LN_EPS 1e-5f

typedef float v2f __attribute__((ext_vector_type(2)));
typedef float v8f __attribute__((ext_vector_type(8)));

// ---------------------------------------------------------------------------
// Kernel 1: segment-mean pooling. batch_idx is sorted, so one block per graph
// binary-searches its [start,end) row range and streams those rows with
// float2 (b64) loads. 256 threads cover the 512 columns. No atomics.
// ---------------------------------------------------------------------------
__device__ __forceinline__ int lower_bound_i64(const long long* __restrict__ a,
                                               int n, long long key) {
  int lo = 0, hi = n;
  while (lo < hi) {
    int mid = (lo + hi) >> 1;
    if (a[mid] < key) lo = mid + 1; else hi = mid;
  }
  return lo;
}

__global__ __launch_bounds__(256)
void pool_kernel(const float* __restrict__ x,
                 const long long* __restrict__ bidx,
                 float* __restrict__ hvn, int n_nodes) {
  const int g = blockIdx.x;
  const int t = threadIdx.x;
  const int start = lower_bound_i64(bidx, n_nodes, (long long)g);
  const int end   = lower_bound_i64(bidx, n_nodes, (long long)g + 1);

  float2 acc = make_float2(0.0f, 0.0f);
  for (int r = start; r < end; ++r) {
    const float2 v = ((const float2*)(x + (size_t)r * DIMH))[t];
    acc.x += v.x;
    acc.y += v.y;
  }
  const float inv = 1.0f / fmaxf((float)(end - start), 1.0f);
  float2* out = (float2*)(hvn + (size_t)g * DIMH);
  out[t] = make_float2(acc.x * inv, acc.y * inv);
}

// ---------------------------------------------------------------------------
// Kernels 2/3: f32 GEMM via V_WMMA_F32_16X16X4_F32.
// One wave -> one 16x16 tile of C; K-loop in steps of 4.
//   A frag (16x4): lane%16 = M;  VGPR0/1 = K {0,1} (lanes 0-15) or {2,3} (16-31)
//   B frag (4x16): lane%16 = N;  same K split
//   C/D frag:      VGPR v, lane l -> row v + 8*(l>=16), col l%16
// ---------------------------------------------------------------------------
template <int K, int N, bool RELU>
__global__ __launch_bounds__(32)
void wmma_gemm_kernel(const float* __restrict__ A,   // [M, K]
                      const float* __restrict__ B,   // [K, N]
                      const float* __restrict__ bias,// [N]
                      float* __restrict__ C) {       // [M, N]
  const int lane = threadIdx.x;
  const int tilesN = N / 16;
  const int tn = blockIdx.x % tilesN;
  const int tm = blockIdx.x / tilesN;

  const int mn = lane & 15;          // M index for A frag / N index for B frag
  const int kh = (lane >> 4) << 1;   // K sub-offset: 0 (lanes 0-15) or 2 (16-31)

  const float* __restrict__ arow = A + (size_t)(tm * 16 + mn) * K;
  const float* __restrict__ bcol = B + (tn * 16 + mn);

  v8f acc = {};
  for (int k = 0; k < K; k += 4) {
    v2f a, b;
    a.x = arow[k + kh];
    a.y = arow[k + kh + 1];
    b.x = bcol[(size_t)(k + kh) * N];
    b.y = bcol[(size_t)(k + kh + 1) * N];
    acc = __builtin_amdgcn_wmma_f32_16x16x4_f32(
        /*neg_a=*/false, a, /*neg_b=*/false, b,
        /*c_mod=*/(short)0, acc, /*reuse_a=*/false, /*reuse_b=*/false);
  }

  const int col  = tn * 16 + mn;
  const int row0 = tm * 16 + ((lane >> 4) << 3);
  const float bv = bias[col];
#pragma unroll
  for (int v = 0; v < 8; ++v) {
    float val = acc[v] + bv;
    if (RELU) val = fmaxf(val, 0.0f);
    C[(size_t)(row0 + v) * N + col] = val;
  }
}

// ---------------------------------------------------------------------------
// Kernel 4: LayerNorm per row of h2 [512,512]; one block per row, LDS reduce.
// Writes the normalized rows directly into the h_vn slot of d_out.
// ---------------------------------------------------------------------------
__global__ __launch_bounds__(256)
void layernorm_kernel(const float* __restrict__ h2,
                      const float* __restrict__ gamma,
                      const float* __restrict__ beta,
                      float* __restrict__ out) {
  const int row = blockIdx.x;
  const int t   = threadIdx.x;
  const float2 v = ((const float2*)(h2 + (size_t)row * DIMH))[t];

  __shared__ float red[256];

  // mean
  red[t] = v.x + v.y;
  __syncthreads();
  for (int off = 128; off > 0; off >>= 1) {
    if (t < off) red[t] += red[t + off];
    __syncthreads();
  }
  const float mu = red[0] * (1.0f / (float)DIMH);
  __syncthreads();

  // variance
  const float dx = v.x - mu, dy = v.y - mu;
  red[t] = dx * dx + dy * dy;
  __syncthreads();
  for (int off = 128; off > 0; off >>= 1) {
    if (t < off) red[t] += red[t + off];
    __syncthreads();
  }
  const float var = red[0] * (1.0f / (float)DIMH);
  const float rs  = rsqrtf(var + LN_EPS);

  const float2 g = ((const float2*)gamma)[t];
  const float2 b = ((const float2*)beta)[t];
  float2* o = (float2*)(out + (size_t)row * DIMH);
  o[t] = make_float2(dx * rs * g.x + b.x, dy * rs * g.y + b.y);
}

// ---------------------------------------------------------------------------
// Kernel 5: residual gather-add. x_new = x + hvn[batch_idx].
// float4 (b128) loads/stores; hvn table is 1 MB -> L2-resident gather.
// ---------------------------------------------------------------------------
__global__ __launch_bounds__(256)
void residual_kernel(const float* __restrict__ x,
                     const long long* __restrict__ bidx,
                     const float* __restrict__ hvn,
                     float* __restrict__ x_new, int n_nodes) {
  const size_t idx = (size_t)blockIdx.x * 256 + threadIdx.x; // over n_nodes*128
  const size_t row = idx >> 7;       // 128 float4 per row
  const int    c4  = (int)(idx & 127);
  if (row >= (size_t)n_nodes) return;

  const long long g = bidx[row];
  const float4 xv = ((const float4*)(x + row * DIMH))[c4];
  const float4 hv = ((const float4*)(hvn + (size_t)g * DIMH))[c4];
  float4 r;
  r.x = xv.x + hv.x;
  r.y = xv.y + hv.y;
  r.z = xv.z + hv.z;
  r.w = xv.w + hv.w;
  ((float4*)(x_new + row * DIMH))[c4] = r;
}

// ---------------------------------------------------------------------------
extern "C" void kernel_launch(void* const* d_in, const int* in_sizes, int n_in,
                              void* d_out, int out_size, void* d_ws, size_t ws_size,
                              hipStream_t stream) {
  const float*     x     = (const float*)d_in[0];
  const long long* bidx  = (const long long*)d_in[1];  // int64 in reference
  const float*     W1    = (const float*)d_in[2];      // [512, 1024]
  const float*     b1    = (const float*)d_in[3];      // [1024]
  const float*     W2    = (const float*)d_in[4];      // [1024, 512]
  const float*     b2    = (const float*)d_in[5];      // [512]
  const float*     gamma = (const float*)d_in[6];      // [512]
  const float*     beta  = (const float*)d_in[7];      // [512]

  const int n_nodes = in_sizes[0] / DIMH;              // 200000

  // workspace layout (floats)
  float* hvn = (float*)d_ws;                           // [512, 512]   1 MB
  float* h1  = hvn + (size_t)NGRAPH * DIMH;            // [512, 1024]  2 MB
  float* h2  = h1  + (size_t)NGRAPH * DIMH2;           // [512, 512]   1 MB

  // output layout: (x_new [n_nodes,512], h_vn [512,512]) concatenated
  float* x_new   = (float*)d_out;
  float* hvn_out = x_new + (size_t)n_nodes * DIMH;

  // 1) segment-mean pooling
  pool_kernel<<<NGRAPH, 256, 0, stream>>>(x, bidx, hvn, n_nodes);

  // 2) h1 = relu(hvn @ W1 + b1)  -- M=512, K=512, N=1024; 2048 wave-tiles
  wmma_gemm_kernel<DIMH, DIMH2, true>
      <<<(NGRAPH / 16) * (DIMH2 / 16), 32, 0, stream>>>(hvn, W1, b1, h1);

  // 3) h2 = h1 @ W2 + b2         -- M=512, K=1024, N=512; 1024 wave-tiles
  wmma_gemm_kernel<DIMH2, DIMH, false>
      <<<(NGRAPH / 16) * (DIMH / 16), 32, 0, stream>>>(h1, W2, b2, h2);

  // 4) LayerNorm -> h_vn output slot of d_out
  layernorm_kernel<<<NGRAPH, 256, 0, stream>>>(h2, gamma, beta, hvn_out);

  // 5) x_new = x + hvn_out[batch_idx]
  const size_t total4 = (size_t)n_nodes * (DIMH / 4);
  residual_kernel<<<(int)((total4 + 255) / 256), 256, 0, stream>>>(
      x, bidx, hvn_out, x_new, n_nodes);
}